// TextureConsistencyLoss_3521873182816
// MI455X (gfx1250) — compile-verified
//
#include <hip/hip_runtime.h>

typedef float v2f __attribute__((ext_vector_type(2)));
typedef float v8f __attribute__((ext_vector_type(8)));

#define PS 8
#define IMG_H 512
#define IMG_W 512
#define IMG_C 3
#define N_PATCH 64          // patches per image
#define PATCH_ELEMS 192     // C*ps*ps
#define GROUP 16            // patches per wave (= WMMA M)
#define TOTAL_PATCHES 2048  // B*N
#define N_GROUPS 128        // TOTAL_PATCHES / GROUP

// One wave (32 lanes) handles 16 gen patches + 16 tgt patches.
// Lane l (l<16): accumulates K=0 (e%4==0) and K=1 (e%4==1) partials for patch l.
// Lane l (l>=16): accumulates K=2,K=3 partials for patch l-16.
// This is exactly the f32 16x4 A-fragment layout, so one
// V_WMMA_F32_16X16X4_F32 vs all-ones B yields all 16 patch totals.
__global__ __launch_bounds__(32) void tcl_patch_stats_kernel(
    const float* __restrict__ gen, const float* __restrict__ tgt,
    const int* __restrict__ yg, const int* __restrict__ xg,
    const int* __restrict__ yt, const int* __restrict__ xt,
    float* __restrict__ partials)
{
    const int g    = blockIdx.x;        // patch group 0..127
    const int lane = threadIdx.x;       // 0..31
    const int m    = lane & 15;         // patch row within group
    const int hi   = lane >> 4;         // 0: K=0,1  1: K=2,3
    const int p    = g * GROUP + m;     // global patch id
    const int b    = p >> 6;            // image index
    const int n    = p & 63;            // patch index within image

    const int ygv = yg[b * N_PATCH + n];
    const int xgv = xg[b * N_PATCH + n];
    const int ytv = yt[b * N_PATCH + n];
    const int xtv = xt[b * N_PATCH + n];

    const float* gbase = gen + (size_t)b * (IMG_C * IMG_H * IMG_W);
    const float* tbase = tgt + (size_t)b * (IMG_C * IMG_H * IMG_W);

    float gs0 = 0.f, gs1 = 0.f, gq0 = 0.f, gq1 = 0.f;
    float ts0 = 0.f, ts1 = 0.f, tq0 = 0.f, tq1 = 0.f;

    // 48 iterations x (2 gen + 2 tgt) floats per lane covers all 192 elements
    // of this lane's patch. e and e+1 always share (c,py) since e%8 is even.
    #pragma unroll 4
    for (int k = 0; k < 48; ++k) {
        const int e   = k * 4 + hi * 2;     // element index within patch
        const int c   = e >> 6;             // channel
        const int rem = e & 63;
        const int py  = rem >> 3;
        const int px  = rem & 7;
        const int gi  = (c * IMG_H + (ygv + py)) * IMG_W + (xgv + px);
        const int ti  = (c * IMG_H + (ytv + py)) * IMG_W + (xtv + px);
        const float a0 = gbase[gi];
        const float a1 = gbase[gi + 1];
        const float c0 = tbase[ti];
        const float c1 = tbase[ti + 1];
        gs0 += a0; gs1 += a1; gq0 += a0 * a0; gq1 += a1 * a1;
        ts0 += c0; ts1 += c1; tq0 += c0 * c0; tq1 += c1 * c1;
    }

    v2f ones; ones.x = 1.0f; ones.y = 1.0f;   // all-ones B (layout-agnostic)
    v8f zacc = {};

    v2f ags; ags.x = gs0; ags.y = gs1;
    v2f agq; agq.x = gq0; agq.y = gq1;
    v2f ats; ats.x = ts0; ats.y = ts1;
    v2f atq; atq.x = tq0; atq.y = tq1;

    // D[M][N] = patch-M total in every column N.
    v8f dgs = __builtin_amdgcn_wmma_f32_16x16x4_f32(false, ags, false, ones, (short)0, zacc, false, false);
    v8f dgq = __builtin_amdgcn_wmma_f32_16x16x4_f32(false, agq, false, ones, (short)0, zacc, false, false);
    v8f dts = __builtin_amdgcn_wmma_f32_16x16x4_f32(false, ats, false, ones, (short)0, zacc, false, false);
    v8f dtq = __builtin_amdgcn_wmma_f32_16x16x4_f32(false, atq, false, ones, (short)0, zacc, false, false);

    // D layout (32-bit C/D 16x16): VGPR r -> M=r (lanes 0-15) / M=8+r (lanes 16-31).
    // Stage per-patch totals through LDS for easy per-patch access.
    __shared__ float Sgs[GROUP], Sgq[GROUP], Sts[GROUP], Stq[GROUP];
    if (lane == 0) {
        #pragma unroll
        for (int r = 0; r < 8; ++r) {
            Sgs[r] = dgs[r]; Sgq[r] = dgq[r]; Sts[r] = dts[r]; Stq[r] = dtq[r];
        }
    } else if (lane == 16) {
        #pragma unroll
        for (int r = 0; r < 8; ++r) {
            Sgs[8 + r] = dgs[r]; Sgq[8 + r] = dgq[r]; Sts[8 + r] = dts[r]; Stq[8 + r] = dtq[r];
        }
    }
    __syncthreads();

    float contrib = 0.0f;
    if (lane < GROUP) {
        const float inv_n   = 1.0f / (float)PATCH_ELEMS;
        const float inv_nm1 = 1.0f / (float)(PATCH_ELEMS - 1);
        const float gsum = Sgs[lane], gss = Sgq[lane];
        const float tsum = Sts[lane], tss = Stq[lane];
        const float gm = gsum * inv_n;
        const float tm = tsum * inv_n;
        const float gv = (gss - gsum * gsum * inv_n) * inv_nm1;  // unbiased var
        const float tv = (tss - tsum * tsum * inv_n) * inv_nm1;
        const float dm = gm - tm;
        const float dv = gv - tv;
        contrib = dm * dm + dv * dv;
    }
    // deterministic wave32 reduction
    #pragma unroll
    for (int off = 16; off > 0; off >>= 1)
        contrib += __shfl_xor(contrib, off, 32);
    if (lane == 0) partials[g] = contrib;
}

// Single-wave deterministic final reduction: fixed strided order + shuffle tree.
__global__ __launch_bounds__(32) void tcl_finalize_kernel(
    const float* __restrict__ partials, float* __restrict__ out)
{
    const int lane = threadIdx.x;
    float s = 0.0f;
    #pragma unroll
    for (int i = 0; i < N_GROUPS / 32; ++i)
        s += partials[lane + i * 32];
    #pragma unroll
    for (int off = 16; off > 0; off >>= 1)
        s += __shfl_xor(s, off, 32);
    if (lane == 0)
        out[0] = s * (1.0f / (float)TOTAL_PATCHES);  // mean_loss + var_loss
}

extern "C" void kernel_launch(void* const* d_in, const int* in_sizes, int n_in,
                              void* d_out, int out_size, void* d_ws, size_t ws_size,
                              hipStream_t stream) {
    const float* gen = (const float*)d_in[0];
    const float* tgt = (const float*)d_in[1];
    const int*   yg  = (const int*)d_in[2];
    const int*   xg  = (const int*)d_in[3];
    const int*   yt  = (const int*)d_in[4];
    const int*   xt  = (const int*)d_in[5];
    float* partials  = (float*)d_ws;          // 128 floats
    float* out       = (float*)d_out;

    tcl_patch_stats_kernel<<<N_GROUPS, 32, 0, stream>>>(gen, tgt, yg, xg, yt, xt, partials);
    tcl_finalize_kernel<<<1, 32, 0, stream>>>(partials, out);
}